// L2LEConv_84859963834438
// MI455X (gfx1250) — compile-verified
//
#include <hip/hip_runtime.h>

typedef float v2f __attribute__((ext_vector_type(2)));
typedef float v4f __attribute__((ext_vector_type(4)));
typedef float v8f __attribute__((ext_vector_type(8)));

#define IN_DIM  128
#define HID_DIM 400
#define OUT_DIM 4
#define NFUSE1  1200   // 3*HID
#define NFUSE2  12     // 3*OUT
#define LDC2    16     // padded row stride for layer-2 abc

static __device__ __forceinline__ int imin(int a, int b) { return a < b ? a : b; }

// ---------------------------------------------------------------------------
// WMMA f32 GEMM:  C[M x N] = A[M x K] * B[K x N] + bias[N]
// Block: 256 threads (8 waves). Tile BM=64 x BN=64, K-chunk KC=32.
// Wave (wm in 0..3, wn in 0..1) computes rows [wm*16,+16) x cols [wn*32,+32)
// as two 16x16 v_wmma_f32_16x16x4_f32 accumulators.
//
// LDS layouts:
//   sA[row][k]   row-major, stride 36: 16B-aligned b128 staging stores,
//                conflict-free 8B fragment reads (36r mod 64 all-distinct).
//   sB K-pair interleaved: element (k,c) at (k>>1)*LDB2 + 2c + (k&1), so a
//                lane's b-fragment {B[k][c], B[k+1][c]} is one aligned 8B
//                read. LDB2 = 160 => lane halves use disjoint bank sets.
// Interior blocks take an unguarded vectorized staging path.
// ---------------------------------------------------------------------------
constexpr int BM = 64, BN = 64, KC = 32;
constexpr int LDA_S = KC + 4;        // 36: padded LDS stride for A rows
constexpr int LDB2  = 2 * BN + 32;   // 160: floats per K-pair row of B

__global__ __launch_bounds__(256)
void gemm_wmma_f32(const float* __restrict__ A, int lda,
                   const float* __restrict__ B, int ldb,
                   const float* __restrict__ bias,
                   float* __restrict__ C, int ldc,
                   int M, int N, int K)
{
    __shared__ float sA[BM * LDA_S];
    __shared__ float sB[(KC / 2) * LDB2];

    const int tid  = threadIdx.x;
    const int lane = tid & 31;
    const int wave = tid >> 5;
    const int wm   = wave & 3;      // row subtile
    const int wn   = wave >> 2;     // 32-col group
    const int ln   = lane & 15;     // m (A) / n (B,C) index within 16
    const int hh   = lane >> 4;     // lane-half: selects K pair / row offset

    const int rowBase = blockIdx.y * BM;
    const int colBase = blockIdx.x * BN;
    const bool fullMN = (rowBase + BM <= M) && (colBase + BN <= N);

    v8f acc0 = {0,0,0,0,0,0,0,0};
    v8f acc1 = {0,0,0,0,0,0,0,0};

    for (int k0 = 0; k0 < K; k0 += KC) {
        __syncthreads();
        if (fullMN && (k0 + KC <= K)) {
            // ---- fast interior staging: unguarded vector loads ----
#pragma unroll
            for (int i = 0; i < 2; ++i) {          // A: 512 float4 total
                int fidx = tid + i * 256;
                int r  = fidx >> 3;                // row 0..63
                int kk = (fidx & 7) * 4;           // k offset 0..28
                v4f v = *(const v4f*)&A[(long)(rowBase + r) * lda + k0 + kk];
                *(v4f*)&sA[r * LDA_S + kk] = v;
            }
#pragma unroll
            for (int i = 0; i < 2; ++i) {          // B: 512 col-pair groups
                int g = tid + i * 256;
                int p = g >> 5;                    // K-pair row 0..15
                int c = (g & 31) * 2;              // col 0..62 (even)
                const float* bp = &B[(long)(k0 + 2 * p) * ldb + colBase + c];
                v2f u = *(const v2f*)bp;           // row 2p  : cols c, c+1
                v2f w = *(const v2f*)(bp + ldb);   // row 2p+1: cols c, c+1
                v4f q; q.x = u.x; q.y = w.x; q.z = u.y; q.w = w.y;
                *(v4f*)&sB[p * LDB2 + c * 2] = q;
            }
        } else {
            // ---- guarded edge-tile staging ----
#pragma unroll
            for (int i = 0; i < (BM * KC) / 256; ++i) {
                int idx = tid + i * 256;
                int r  = idx >> 5;
                int kk = idx & (KC - 1);
                int gr = rowBase + r, gk = k0 + kk;
                float v = 0.f;
                if (gr < M && gk < K) v = A[(long)gr * lda + gk];
                sA[r * LDA_S + kk] = v;
            }
#pragma unroll
            for (int i = 0; i < (KC * BN) / 256; ++i) {
                int idx = tid + i * 256;
                int r  = idx >> 6;
                int cc = idx & (BN - 1);
                int gk = k0 + r, gc = colBase + cc;
                float v = 0.f;
                if (gk < K && gc < N) v = B[(long)gk * ldb + gc];
                sB[(r >> 1) * LDB2 + cc * 2 + (r & 1)] = v;
            }
        }
        __syncthreads();

#pragma unroll
        for (int kk = 0; kk < KC; kk += 4) {
            // A frag 16x4: lane half hh holds K = kk+2hh, kk+2hh+1 for row ln
            int ar = (wm * 16 + ln) * LDA_S + kk + 2 * hh;   // even => 8B aligned
            v2f a = *(const v2f*)&sA[ar];
            // B frag 4x16: contiguous {B[k][c], B[k+1][c]} pair per lane
            int bp = ((kk >> 1) + hh) * LDB2 + (wn * 32 + ln) * 2;
            v2f b0 = *(const v2f*)&sB[bp];
            v2f b1 = *(const v2f*)&sB[bp + 32];
            acc0 = __builtin_amdgcn_wmma_f32_16x16x4_f32(
                       false, a, false, b0, (short)0, acc0, false, false);
            acc1 = __builtin_amdgcn_wmma_f32_16x16x4_f32(
                       false, a, false, b1, (short)0, acc1, false, false);
        }
    }

    // C/D layout: VGPR r -> row r (lanes 0-15) / row r+8 (lanes 16-31), col=ln
    if (fullMN) {
#pragma unroll
        for (int r = 0; r < 8; ++r) {
            int row = rowBase + wm * 16 + r + 8 * hh;
            int c0  = colBase + wn * 32 + ln;
            C[(long)row * ldc + c0]      = acc0[r] + bias[c0];
            C[(long)row * ldc + c0 + 16] = acc1[r] + bias[c0 + 16];
        }
    } else {
#pragma unroll
        for (int r = 0; r < 8; ++r) {
            int row = rowBase + wm * 16 + r + 8 * hh;
            int c0  = colBase + wn * 32 + ln;
            int c1  = c0 + 16;
            if (row < M) {
                if (c0 < N) C[(long)row * ldc + c0] = acc0[r] + bias[c0];
                if (c1 < N) C[(long)row * ldc + c1] = acc1[r] + bias[c1];
            }
        }
    }
}

// ---------------------------------------------------------------------------
// Weight packing: Wf1 = [W1a | W2a | W3a] (128 x 1200), biasf1 = [b1a|0|b3a]
// ---------------------------------------------------------------------------
__global__ void pack_w1(const float* __restrict__ W1, const float* __restrict__ b1,
                        const float* __restrict__ W2, const float* __restrict__ W3,
                        const float* __restrict__ b3,
                        float* __restrict__ Wf, float* __restrict__ biasf)
{
    int idx = blockIdx.x * blockDim.x + threadIdx.x;
    if (idx < IN_DIM * NFUSE1) {
        int k = idx / NFUSE1, c = idx % NFUSE1;
        float v;
        if (c < HID_DIM)            v = W1[k * HID_DIM + c];
        else if (c < 2 * HID_DIM)   v = W2[k * HID_DIM + c - HID_DIM];
        else                        v = W3[k * HID_DIM + c - 2 * HID_DIM];
        Wf[idx] = v;
    }
    if (idx < NFUSE1) {
        float bv = (idx < HID_DIM) ? b1[idx]
                 : (idx < 2 * HID_DIM) ? 0.f
                 : b3[idx - 2 * HID_DIM];
        biasf[idx] = bv;
    }
}

__global__ void pack_w2(const float* __restrict__ W1, const float* __restrict__ b1,
                        const float* __restrict__ W2, const float* __restrict__ W3,
                        const float* __restrict__ b3,
                        float* __restrict__ Wf, float* __restrict__ biasf)
{
    int idx = blockIdx.x * blockDim.x + threadIdx.x;
    if (idx < HID_DIM * NFUSE2) {
        int k = idx / NFUSE2, c = idx % NFUSE2;
        int g = c >> 2, j = c & 3;
        const float* W = (g == 0) ? W1 : (g == 1) ? W2 : W3;
        Wf[idx] = W[k * OUT_DIM + j];
    }
    if (idx < NFUSE2) {
        int g = idx >> 2, j = idx & 3;
        biasf[idx] = (g == 0) ? b1[j] : (g == 1) ? 0.f : b3[j];
    }
}

// ---------------------------------------------------------------------------
// CSR construction
// ---------------------------------------------------------------------------
__global__ void zero_i32(int* __restrict__ p, int n)
{
    int i = blockIdx.x * blockDim.x + threadIdx.x;
    if (i < n) p[i] = 0;
}

__global__ void count_deg(const int* __restrict__ dst, int* __restrict__ counts, int E)
{
    int e = blockIdx.x * blockDim.x + threadIdx.x;
    if (e < E) atomicAdd(&counts[dst[e]], 1);
}

__global__ __launch_bounds__(1024)
void scan_build(const int* __restrict__ counts, int* __restrict__ rowp,
                int* __restrict__ cursor, float* __restrict__ deg, int n)
{
    __shared__ int lsum[1024];
    int t = threadIdx.x;
    int chunk = (n + 1023) / 1024;
    int beg = t * chunk;
    int end = imin(beg + chunk, n);
    int s = 0;
    for (int i = beg; i < end; ++i) s += counts[i];
    lsum[t] = s;
    __syncthreads();
    for (int off = 1; off < 1024; off <<= 1) {
        int v = (t >= off) ? lsum[t - off] : 0;
        __syncthreads();
        lsum[t] += v;
        __syncthreads();
    }
    int run = lsum[t] - s;                      // exclusive prefix
    for (int i = beg; i < end; ++i) {
        rowp[i]   = run;
        cursor[i] = run;
        deg[i]    = (float)counts[i];
        run += counts[i];
    }
    if (end == n) rowp[n] = run;                // total (benign dup writes)
}

__global__ void fill_csr(const int* __restrict__ src, const int* __restrict__ dst,
                         int* __restrict__ cursor, int* __restrict__ csr, int E)
{
    int e = blockIdx.x * blockDim.x + threadIdx.x;
    if (e < E) {
        int d   = dst[e];
        int pos = atomicAdd(&cursor[d], 1);
        csr[pos] = src[e];
    }
}

// ---------------------------------------------------------------------------
// Layer-1 aggregation + fused epilogue: h = relu(sum_j a1[j] - deg*b1 + c1)
// One block per node; 512 threads cover the 400 channels; edge list staged
// through LDS in chunks so all lanes gather coalesced rows of a1.
// ---------------------------------------------------------------------------
__global__ __launch_bounds__(512)
void agg1_relu(const float* __restrict__ abc, const int* __restrict__ rowp,
               const int* __restrict__ csrc, const float* __restrict__ deg,
               float* __restrict__ h)
{
    __shared__ int ssrc[512];
    const int node = blockIdx.x;
    const int t    = threadIdx.x;
    const int beg  = rowp[node];
    const int end  = rowp[node + 1];

    float acc = 0.f;
    for (int base = beg; base < end; base += 512) {
        int cnt = imin(512, end - base);
        __syncthreads();
        if (t < cnt) ssrc[t] = csrc[base + t];
        __syncthreads();
        if (t < HID_DIM) {
            for (int j = 0; j < cnt; ++j)
                acc += abc[(long)ssrc[j] * NFUSE1 + t];     // 'a' block: cols 0..399
        }
    }
    if (t < HID_DIM) {
        float dg = deg[node];
        long  off = (long)node * NFUSE1;
        float v = acc - dg * abc[off + HID_DIM + t] + abc[off + 2 * HID_DIM + t];
        h[(long)node * HID_DIM + t] = v > 0.f ? v : 0.f;
    }
}

// ---------------------------------------------------------------------------
// Layer-2 aggregation + output: out = relu(sum_j a2[j] - deg*b2 + c2)
// ---------------------------------------------------------------------------
__global__ void agg2_out(const float* __restrict__ abc2, const int* __restrict__ rowp,
                         const int* __restrict__ csrc, const float* __restrict__ deg,
                         float* __restrict__ out, int n)
{
    int gid  = blockIdx.x * blockDim.x + threadIdx.x;
    int node = gid >> 2;
    int c    = gid & 3;
    if (node >= n) return;
    int beg = rowp[node], end = rowp[node + 1];
    float acc = 0.f;
    for (int e = beg; e < end; ++e)
        acc += abc2[csrc[e] * LDC2 + c];
    float v = acc - deg[node] * abc2[node * LDC2 + OUT_DIM + c]
                  + abc2[node * LDC2 + 2 * OUT_DIM + c];
    out[node * OUT_DIM + c] = v > 0.f ? v : 0.f;
}

// ---------------------------------------------------------------------------
extern "C" void kernel_launch(void* const* d_in, const int* in_sizes, int n_in,
                              void* d_out, int out_size, void* d_ws, size_t ws_size,
                              hipStream_t stream)
{
    const float* x   = (const float*)d_in[0];
    const int*   ei  = (const int*)d_in[1];
    const float* W1a = (const float*)d_in[2];
    const float* b1a = (const float*)d_in[3];
    const float* W2a = (const float*)d_in[4];
    const float* W3a = (const float*)d_in[5];
    const float* b3a = (const float*)d_in[6];
    const float* W1b = (const float*)d_in[7];
    const float* b1b = (const float*)d_in[8];
    const float* W2b = (const float*)d_in[9];
    const float* W3b = (const float*)d_in[10];
    const float* b3b = (const float*)d_in[11];

    const int n = in_sizes[0] / IN_DIM;   // 10000 nodes
    const int E = in_sizes[1] / 2;        // 320000 edges
    const int* src = ei;
    const int* dst = ei + E;

    // workspace bump allocator (~67 MB total)
    char* ws = (char*)d_ws;
    auto alloc = [&](size_t bytes) -> char* {
        char* p = ws;
        ws += (bytes + 255) & ~size_t(255);
        return p;
    };
    float* Wf1  = (float*)alloc(sizeof(float) * IN_DIM * NFUSE1);
    float* bf1  = (float*)alloc(sizeof(float) * NFUSE1);
    float* Wf2  = (float*)alloc(sizeof(float) * HID_DIM * NFUSE2);
    float* bf2  = (float*)alloc(sizeof(float) * NFUSE2);
    float* abc1 = (float*)alloc(sizeof(float) * (size_t)n * NFUSE1);
    float* h    = (float*)alloc(sizeof(float) * (size_t)n * HID_DIM);
    float* abc2 = (float*)alloc(sizeof(float) * (size_t)n * LDC2);
    int*   counts = (int*)alloc(sizeof(int) * n);
    int*   rowp   = (int*)alloc(sizeof(int) * (n + 1));
    int*   cursor = (int*)alloc(sizeof(int) * n);
    float* deg    = (float*)alloc(sizeof(float) * n);
    int*   csr    = (int*)alloc(sizeof(int) * E);

    // weight packing
    pack_w1<<<(IN_DIM * NFUSE1 + 255) / 256, 256, 0, stream>>>(W1a, b1a, W2a, W3a, b3a, Wf1, bf1);
    pack_w2<<<(HID_DIM * NFUSE2 + 255) / 256, 256, 0, stream>>>(W1b, b1b, W2b, W3b, b3b, Wf2, bf2);

    // CSR build (also yields degrees)
    zero_i32<<<(n + 255) / 256, 256, 0, stream>>>(counts, n);
    count_deg<<<(E + 255) / 256, 256, 0, stream>>>(dst, counts, E);
    scan_build<<<1, 1024, 0, stream>>>(counts, rowp, cursor, deg, n);
    fill_csr<<<(E + 255) / 256, 256, 0, stream>>>(src, dst, cursor, csr, E);

    // layer 1: abc1 = x @ [W1a|W2a|W3a] + [b1a|0|b3a]   (WMMA f32)
    dim3 g1((NFUSE1 + BN - 1) / BN, (n + BM - 1) / BM);
    gemm_wmma_f32<<<g1, 256, 0, stream>>>(x, IN_DIM, Wf1, NFUSE1, bf1,
                                          abc1, NFUSE1, n, NFUSE1, IN_DIM);

    agg1_relu<<<n, 512, 0, stream>>>(abc1, rowp, csr, deg, h);

    // layer 2: abc2 = h @ [W1b|W2b|W3b] + [b1b|0|b3b]   (WMMA f32, N=12, ldc=16)
    dim3 g2((NFUSE2 + BN - 1) / BN, (n + BM - 1) / BM);
    gemm_wmma_f32<<<g2, 256, 0, stream>>>(h, HID_DIM, Wf2, NFUSE2, bf2,
                                          abc2, LDC2, n, NFUSE2, HID_DIM);

    agg2_out<<<(n * OUT_DIM + 255) / 256, 256, 0, stream>>>(abc2, rowp, csr, deg,
                                                            (float*)d_out, n);
}